// SmoothTopkSVM_2010044695269
// MI455X (gfx1250) — compile-verified
//
#include <hip/hip_runtime.h>
#include <hip/hip_bf16.h>
#include <stdint.h>

#define NCLS   1000
#define VEC4S  250            // NCLS / 4
#define WPB    8              // waves per block
#define BLKT   256
#define NBLOCKS 2048
// k*tau*log(thresh) = 5*log(1000)
#define HARD_THRESH 34.538776394910684f

__device__ __forceinline__ float logaddexpf_(float a, float b) {
    float mx = fmaxf(a, b);
    float d  = fminf(a, b) - mx;          // <= 0
    return mx + log1pf(__expf(d));
}

__device__ __forceinline__ void ins6(float t[6], float v) {
    if (v > t[5]) {
        t[5] = v;
#pragma unroll
        for (int i = 5; i > 0; --i) {
            float hi = t[i - 1], lo = t[i];
            if (lo > hi) { t[i - 1] = lo; t[i] = hi; }
        }
    }
}
__device__ __forceinline__ void ins5(float t[5], float v) {
    if (v > t[4]) {
        t[4] = v;
#pragma unroll
        for (int i = 4; i > 0; --i) {
            float hi = t[i - 1], lo = t[i];
            if (lo > hi) { t[i - 1] = lo; t[i] = hi; }
        }
    }
}

// Issue one row (250 x b128) as async global->LDS copies. 8 instructions/wave.
__device__ __forceinline__ void issue_row(const float* __restrict__ src,
                                          float* ldsDst, int lane) {
#pragma unroll
    for (int j = 0; j < 8; ++j) {
        int idx = j * 32 + lane;
        if (idx < VEC4S) {
            unsigned lAddr = (unsigned)(uintptr_t)(ldsDst + idx * 4); // LDS byte offset
            unsigned long long g = (unsigned long long)(uintptr_t)(src + idx * 4);
            asm volatile("global_load_async_to_lds_b128 %0, %1, off"
                         :: "v"(lAddr), "v"(g) : "memory");
        }
    }
}

__global__ __launch_bounds__(BLKT)
void topk_svm_kernel(const float* __restrict__ x, const int* __restrict__ y,
                     float* __restrict__ partial, int nrows) {
    __shared__ float lds[WPB * 2 * 1024];   // 8 waves x 2 buffers x 1024 floats (64 KB)
    __shared__ float red[WPB];

    const int lane = threadIdx.x & 31;
    const int wv   = threadIdx.x >> 5;
    const int totalWaves = gridDim.x * WPB;

    float* buf0 = &lds[(wv * 2 + 0) * 1024];
    float* buf1 = &lds[(wv * 2 + 1) * 1024];

    int   row   = blockIdx.x * WPB + wv;
    int   buf   = 0;
    bool  valid = (row < nrows);
    float waveSum = 0.0f;

    if (valid) issue_row(x + (size_t)row * NCLS, buf0, lane);

    while (valid) {
        int  nrow = row + totalWaves;
        bool pre  = (nrow < nrows);
        float* cur = buf ? buf1 : buf0;
        float* nxt = buf ? buf0 : buf1;

        if (pre) {
            issue_row(x + (size_t)nrow * NCLS, nxt, lane);
            asm volatile("s_wait_asynccnt 0x8" ::: "memory");  // current row's 8 done
        } else {
            asm volatile("s_wait_asynccnt 0x0" ::: "memory");
        }

        const int yv = y[row];

        // ---- pass A: per-lane top-6 of x, top-5 of x1 ----
        float top6x[6], top5n[5];
#pragma unroll
        for (int i = 0; i < 6; ++i) top6x[i] = -3.0e38f;
#pragma unroll
        for (int i = 0; i < 5; ++i) top5n[i] = -3.0e38f;

#pragma unroll
        for (int j = 0; j < 8; ++j) {
            int idx = j * 32 + lane;
            if (idx < VEC4S) {
                float4 v = *reinterpret_cast<const float4*>(cur + idx * 4);
                int e = idx * 4;
                float vv[4] = {v.x, v.y, v.z, v.w};
#pragma unroll
                for (int c = 0; c < 4; ++c) {
                    float val = vv[c];
                    ins6(top6x, val);
                    float v1 = ((e + c) == yv) ? -3.0e38f : val;
                    ins5(top5n, v1);
                }
            }
        }
        // butterfly merge (snapshot partner BEFORE inserting: keeps sets disjoint)
#pragma unroll
        for (int m = 1; m < 32; m <<= 1) {
            float o6[6], o5[5];
#pragma unroll
            for (int t = 0; t < 6; ++t) o6[t] = __shfl_xor(top6x[t], m, 32);
#pragma unroll
            for (int t = 0; t < 5; ++t) o5[t] = __shfl_xor(top5n[t], m, 32);
#pragma unroll
            for (int t = 0; t < 6; ++t) ins6(top6x, o6[t]);
#pragma unroll
            for (int t = 0; t < 5; ++t) ins5(top5n, o5[t]);
        }
        const float s = top5n[0] * 0.2f;   // max of a = x1/5 (shift)

        // ---- pass B: shifted linear-domain ESP e0..e5 ----
        float E[6] = {1.f, 0.f, 0.f, 0.f, 0.f, 0.f};
#pragma unroll
        for (int j = 0; j < 8; ++j) {
            int idx = j * 32 + lane;
            if (idx < VEC4S) {
                float4 v = *reinterpret_cast<const float4*>(cur + idx * 4);
                int e = idx * 4;
                float vv[4] = {v.x, v.y, v.z, v.w};
#pragma unroll
                for (int c = 0; c < 4; ++c) {
                    float t = ((e + c) == yv) ? 0.0f : __expf(vv[c] * 0.2f - s);
#pragma unroll
                    for (int mdeg = 5; mdeg >= 1; --mdeg)
                        E[mdeg] = fmaf(E[mdeg - 1], t, E[mdeg]);
                }
            }
        }
        // butterfly merge: truncated polynomial product over disjoint lane sets
#pragma unroll
        for (int m = 1; m < 32; m <<= 1) {
            float o[6];
#pragma unroll
            for (int t = 0; t < 6; ++t) o[t] = __shfl_xor(E[t], m, 32);
            float ne[6];
#pragma unroll
            for (int mm = 0; mm < 6; ++mm) {
                float acc = 0.f;
#pragma unroll
                for (int j = 0; j <= mm; ++j) acc = fmaf(E[j], o[mm - j], acc);
                ne[mm] = acc;
            }
#pragma unroll
            for (int t = 0; t < 6; ++t) E[t] = ne[t];
        }

        const float x2       = cur[yv];
        const float log_ek   = __logf(E[5]) + 5.0f * s;
        const float log_ekm1 = __logf(E[4]) + 4.0f * s;
        const float loss_neg = log_ekm1 + x2 * 0.2f;
        const float loss_pos = logaddexpf_(log_ek + 1.0f, loss_neg); // alpha/tau = 1
        const float l_s      = loss_pos - loss_neg;                  // tau = 1

        const float sum5 = top5n[0] + top5n[1] + top5n[2] + top5n[3] + top5n[4];
        const float max1 = sum5 * 0.2f + 1.0f;                       // mean(top5)+alpha
        const float max2 = (sum5 - top5n[4] + x2) * 0.2f;            // (top4sum+x2)/5
        const float l_h  = fmaxf(max1 - max2, 0.0f);

        const bool hard = (top6x[4] - top6x[5]) >= HARD_THRESH;
        waveSum += hard ? l_h : l_s;

        row = nrow; buf ^= 1; valid = pre;
    }

    if (lane == 0) red[wv] = waveSum;
    __syncthreads();
    if (threadIdx.x == 0) {
        float b = 0.f;
#pragma unroll
        for (int i = 0; i < WPB; ++i) b += red[i];
        partial[blockIdx.x] = b;
    }
}

__global__ __launch_bounds__(256)
void finalize_kernel(const float* __restrict__ partial, int nparts,
                     float* __restrict__ out, float invN) {
    __shared__ float smem[256];
    float acc = 0.f;
    for (int i = threadIdx.x; i < nparts; i += 256) acc += partial[i];
    smem[threadIdx.x] = acc;
    __syncthreads();
    for (int off = 128; off > 0; off >>= 1) {
        if (threadIdx.x < (unsigned)off) smem[threadIdx.x] += smem[threadIdx.x + off];
        __syncthreads();
    }
    if (threadIdx.x == 0) out[0] = smem[0] * invN;
}

extern "C" void kernel_launch(void* const* d_in, const int* in_sizes, int n_in,
                              void* d_out, int out_size, void* d_ws, size_t ws_size,
                              hipStream_t stream) {
    const float* x = (const float*)d_in[0];
    const int*   y = (const int*)d_in[1];
    const int nrows = in_sizes[1];          // 32768
    float* partial = (float*)d_ws;          // NBLOCKS floats

    topk_svm_kernel<<<NBLOCKS, BLKT, 0, stream>>>(x, y, partial, nrows);
    finalize_kernel<<<1, 256, 0, stream>>>(partial, NBLOCKS, (float*)d_out,
                                           1.0f / (float)nrows);
}